// MultiBoxLoss_71519795413464
// MI455X (gfx1250) — compile-verified
//
#include <hip/hip_runtime.h>
#include <math.h>

// Problem constants (match reference)
constexpr int NB = 64;      // batch
constexpr int NP = 16800;   // priors
constexpr int NP4 = NP / 4; // float4 chunks per row (16800 % 4 == 0)
constexpr int NG = 32;      // ground-truth boxes
constexpr float VAR0 = 0.1f;
constexpr float VAR1 = 0.2f;
constexpr float THRESH = 0.35f;
constexpr int NEGPOS = 7;

typedef __attribute__((ext_vector_type(2))) float v2f;
typedef __attribute__((ext_vector_type(8))) float v8f;

struct Acc {
    float loss_l;
    float loss_c;
    float loss_landm;
    int   npos_total;
    int   nposlm_total;
    int   npos[NB];
};

// ---------------------------------------------------------------------------
// Async DMA: global -> LDS, 16 bytes per lane, tracked by ASYNCcnt.
// lds_off is the byte offset inside the wave's LDS allocation (low 32 bits of
// the generic shared-pointer address, per ISA 10.2 LDS aperture mapping).
// ---------------------------------------------------------------------------
__device__ __forceinline__ void async_load16(unsigned lds_off, const void* gaddr) {
    asm volatile("global_load_async_to_lds_b128 %0, %1, off"
                 :: "v"(lds_off), "v"((unsigned long long)(uintptr_t)gaddr)
                 : "memory");
}
__device__ __forceinline__ void wait_async0() {
    asm volatile("s_wait_asynccnt 0x0" ::: "memory");
}
__device__ __forceinline__ unsigned lds_offset_of(const void* shared_ptr) {
    return (unsigned)(uintptr_t)shared_ptr;  // addr[31:0] == LDS offset
}

// ---------------------------------------------------------------------------
// Exact wave32 sum via V_WMMA_F32_16X16X4_F32.
// A is 16x4: lane L<16 holds A[L][0]=v (K=0 slot), lane L>=16 holds A[L-16][2]=v.
// With B = all-ones 4x16 and C = 0:  D[m][n] = v(lane m) + v(lane m+16).
// Lane 0 holds D[0..7][0] and lane 16 holds D[8..15][0], so
// total = sum_j d[j]@lane0 + sum_j d[j]@lane16  (combined with v_readlane).
// EXEC must be all ones at the call site (callers guarantee convergence).
// ---------------------------------------------------------------------------
__device__ __forceinline__ float wave_sum_wmma(float v) {
    v2f a; a.x = v;    a.y = 0.0f;
    v2f b; b.x = 1.0f; b.y = 1.0f;
    v8f c = {};
    v8f d = __builtin_amdgcn_wmma_f32_16x16x4_f32(false, a, false, b,
                                                  (short)0, c, false, false);
    float s = d[0] + d[1] + d[2] + d[3] + d[4] + d[5] + d[6] + d[7];
    int si = __float_as_int(s);
    return __int_as_float(__builtin_amdgcn_readlane(si, 0)) +
           __int_as_float(__builtin_amdgcn_readlane(si, 16));
}

__device__ __forceinline__ float smooth_l1(float x) {
    float a = fabsf(x);
    return (a < 1.0f) ? 0.5f * a * a : a - 0.5f;
}

// ---------------------------------------------------------------------------
// Init: zero per-truth argmax keys and accumulators.
// ---------------------------------------------------------------------------
__global__ __launch_bounds__(256) void k_init(unsigned long long* bp, Acc* acc) {
    int i = blockIdx.x * 256 + threadIdx.x;
    if (i < NB * NG) bp[i] = 0ull;
    if (i == 0) {
        acc->loss_l = 0.0f; acc->loss_c = 0.0f; acc->loss_landm = 0.0f;
        acc->npos_total = 0; acc->nposlm_total = 0;
    }
    if (i < NB) acc->npos[i] = 0;
}

// ---------------------------------------------------------------------------
// Kernel 1: per (b,p): IoU against all 32 truths.
//   - best over g (value + first-max index)  -> bto / bti
//   - per-truth best prior: pack (iou_bits << 32) | (~p) and 64-bit max-reduce
//     (wave shuffle -> LDS atomicMax -> global atomicMax), giving exact
//     argmax with smallest-index tie-break.
// Targets block (32x15 floats = 1920B) staged into LDS via async DMA.
// ---------------------------------------------------------------------------
__global__ __launch_bounds__(256) void k_match(
    const float* __restrict__ priors, const float* __restrict__ targets,
    float* __restrict__ bto, int* __restrict__ bti, int* __restrict__ assigned,
    unsigned long long* __restrict__ bp)
{
    __shared__ float sTG[NG * 15];            // raw targets block
    __shared__ unsigned long long sBP[NG];

    const int tid = threadIdx.x;
    const int b   = blockIdx.x;
    const int p   = blockIdx.y * 256 + tid;

    {
        const char* tg = (const char*)(targets + (size_t)b * NG * 15);
        unsigned base = lds_offset_of((const void*)sTG);
        if (tid < (NG * 15 * 4) / 16)         // 120 x 16B = 1920B
            async_load16(base + tid * 16, tg + (size_t)tid * 16);
    }
    if (tid < NG) sBP[tid] = 0ull;
    wait_async0();
    __syncthreads();

    const bool act = (p < NP);
    float px1 = 0.f, py1 = 0.f, px2 = 0.f, py2 = 0.f, areaB = 0.f;
    if (act) {
        const float4 pr = *(const float4*)(priors + (size_t)p * 4);
        px1 = pr.x - pr.z * 0.5f; py1 = pr.y - pr.w * 0.5f;
        px2 = pr.x + pr.z * 0.5f; py2 = pr.y + pr.w * 0.5f;
        areaB = pr.z * pr.w;
    }

    float best = -1.0f;
    int   besti = 0;
    const int lane = tid & 31;

    for (int g = 0; g < NG; ++g) {
        unsigned long long key = 0ull;
        float x1 = sTG[g * 15 + 0], y1 = sTG[g * 15 + 1];
        float x2 = sTG[g * 15 + 2], y2 = sTG[g * 15 + 3];
        if (act) {
            float ix = fminf(px2, x2) - fmaxf(px1, x1);
            float iy = fminf(py2, y2) - fmaxf(py1, y1);
            ix = fmaxf(ix, 0.0f); iy = fmaxf(iy, 0.0f);
            float inter = ix * iy;
            float areaA = (x2 - x1) * (y2 - y1);
            float iou = inter / (areaA + areaB - inter);
            if (iou > best) { best = iou; besti = g; }   // first-max over g
            key = ((unsigned long long)__float_as_uint(iou) << 32) |
                  (unsigned)(0xFFFFFFFFu - (unsigned)p);  // smaller p wins ties
        }
        // wave32 max-reduce of packed key
        for (int off = 16; off > 0; off >>= 1) {
            unsigned long long o = __shfl_xor(key, off, 32);
            key = (key > o) ? key : o;
        }
        if (lane == 0) atomicMax(&sBP[g], key);
    }

    if (act) {
        size_t idx = (size_t)b * NP + p;
        bto[idx]      = best;
        bti[idx]      = besti;
        assigned[idx] = -1;
    }
    __syncthreads();
    if (tid < NG) atomicMax(&bp[(size_t)b * NG + tid], sBP[tid]);
}

// ---------------------------------------------------------------------------
// Kernel 2: per (b,g) scatter corrections (one wave32 per batch):
//   assigned[best_prior_idx] = max(g)        (all g)
//   bto[best_prior_idx]      = max(2.0)      (valid g only)
//   anyvalid[b] = any(best_prior_overlap >= 0.2)
// ---------------------------------------------------------------------------
__global__ __launch_bounds__(32) void k_bestprior(
    const unsigned long long* __restrict__ bp,
    float* __restrict__ bto, int* __restrict__ assigned, int* __restrict__ anyvalid)
{
    const int b = blockIdx.x;
    const int g = threadIdx.x;
    unsigned long long key = bp[(size_t)b * NG + g];
    float ov  = __uint_as_float((unsigned)(key >> 32));
    int   pix = (int)(0xFFFFFFFFu - (unsigned)(key & 0xFFFFFFFFu));
    bool valid = (ov >= 0.2f);
    unsigned long long bal = __ballot(valid);
    if (g == 0) anyvalid[b] = (bal != 0ull) ? 1 : 0;
    size_t idx = (size_t)b * NP + pix;
    atomicMax(&assigned[idx], g);
    if (valid) atomicMax((unsigned*)&bto[idx], __float_as_uint(2.0f)); // bto >= 0
}

// ---------------------------------------------------------------------------
// Kernel 3: per (b,p): conf target, loc/landm encode, smooth-L1 partials,
// per-prior loss_c (pos rows zeroed), pos counts. Float sums reduced per-wave
// with WMMA, combined in LDS, one atomicAdd per block per quantity.
// Targets block async-DMA'd into LDS.
// ---------------------------------------------------------------------------
__global__ __launch_bounds__(256) void k_encode_loss(
    const float* __restrict__ loc_data, const float* __restrict__ conf_data,
    const float* __restrict__ landm_data, const float* __restrict__ priors,
    const float* __restrict__ targets,
    const float* __restrict__ bto, const int* __restrict__ bti,
    const int* __restrict__ assigned, const int* __restrict__ anyvalid,
    float* __restrict__ lossc, Acc* __restrict__ acc)
{
    __shared__ float sTG[NG * 15];   // [g*15+0..3]=box, +4..13=landm, +14=label
    __shared__ float sSum[3][8];
    __shared__ int   sNp[2];

    const int tid = threadIdx.x;
    const int b   = blockIdx.x;
    const int p   = blockIdx.y * 256 + tid;

    {
        const char* tg = (const char*)(targets + (size_t)b * NG * 15);
        unsigned base = lds_offset_of((const void*)sTG);
        if (tid < (NG * 15 * 4) / 16)
            async_load16(base + tid * 16, tg + (size_t)tid * 16);
    }
    if (tid < 2) sNp[tid] = 0;
    wait_async0();
    __syncthreads();

    float ll = 0.0f, llm = 0.0f, pce = 0.0f;
    int isPos = 0, isPos1 = 0;

    if (p < NP) {
        size_t idx = (size_t)b * NP + p;
        int anyv = anyvalid[b];
        int a    = assigned[idx];
        int tix  = (a >= 0) ? a : bti[idx];
        float ov = bto[idx];
        int conf = 0;
        if (anyv) conf = (ov < THRESH) ? 0 : (int)sTG[tix * 15 + 14];
        bool pos  = (conf != 0);
        bool pos1 = (conf > 0);
        isPos = pos; isPos1 = pos1;

        const float4 pr = *(const float4*)(priors + (size_t)p * 4);

        if (pos) {
            float x1 = sTG[tix * 15 + 0], y1 = sTG[tix * 15 + 1];
            float x2 = sTG[tix * 15 + 2], y2 = sTG[tix * 15 + 3];
            float gx = ((x1 + x2) * 0.5f - pr.x) / (VAR0 * pr.z);
            float gy = ((y1 + y2) * 0.5f - pr.y) / (VAR0 * pr.w);
            float gw = logf((x2 - x1) / pr.z) / VAR1;
            float gh = logf((y2 - y1) / pr.w) / VAR1;
            const float* ld = loc_data + idx * 4;
            ll = smooth_l1(ld[0] - gx) + smooth_l1(ld[1] - gy) +
                 smooth_l1(ld[2] - gw) + smooth_l1(ld[3] - gh);
        }
        if (pos1) {
            const float* lm = landm_data + idx * 10;
            __builtin_prefetch((const void*)(lm + 10 * 256), 0, 1); // global_prefetch_b8
            #pragma unroll
            for (int j = 0; j < 5; ++j) {
                float tx = (sTG[tix * 15 + 4 + 2 * j]     - pr.x) / (VAR0 * pr.z);
                float ty = (sTG[tix * 15 + 4 + 2 * j + 1] - pr.y) / (VAR0 * pr.w);
                llm += smooth_l1(lm[2 * j] - tx) + smooth_l1(lm[2 * j + 1] - ty);
            }
        }
        const float* cd = conf_data + idx * 2;
        __builtin_prefetch((const void*)(cd + 2 * 256), 0, 1);
        float c0 = cd[0], c1 = cd[1];
        float m  = fmaxf(c0, c1);
        float lse = m + logf(expf(c0 - m) + expf(c1 - m));
        float lc;
        if (pos) { lc = 0.0f; pce = lse - c1; }  // CE at class 1 for positives
        else     { lc = lse - c0; }              // mining score for negatives
        lossc[idx] = lc;
    }

    // Converged: EXEC all ones for WMMA reductions.
    float s0 = wave_sum_wmma(ll);
    float s1 = wave_sum_wmma(llm);
    float s2 = wave_sum_wmma(pce);
    const int wave = tid >> 5, lane = tid & 31;
    if (lane == 0) { sSum[0][wave] = s0; sSum[1][wave] = s1; sSum[2][wave] = s2; }
    if (isPos)  atomicAdd(&sNp[0], 1);
    if (isPos1) atomicAdd(&sNp[1], 1);
    __syncthreads();

    if (tid == 0) {
        float t0 = 0.f, t1 = 0.f, t2 = 0.f;
        #pragma unroll
        for (int w = 0; w < 8; ++w) { t0 += sSum[0][w]; t1 += sSum[1][w]; t2 += sSum[2][w]; }
        atomicAdd(&acc->loss_l,     t0);
        atomicAdd(&acc->loss_landm, t1);
        atomicAdd(&acc->loss_c,     t2);  // positive-row CE part
        int np = sNp[0], np1 = sNp[1];
        if (np)  { atomicAdd(&acc->npos[b], np); atomicAdd(&acc->npos_total, np); }
        if (np1) atomicAdd(&acc->nposlm_total, np1);
    }
}

// ---------------------------------------------------------------------------
// Kernel 4: hard negative mining, one block per batch row.
// The 67 KB loss_c row is DMA'd once into LDS (GLOBAL_LOAD_ASYNC_TO_LDS_B128,
// overlapped with histogram zeroing), then a 3-pass radix select (11+11+10
// bits) over float bit patterns (all values >= 0 -> unsigned order == float
// order) finds t = k-th largest. Selected-set sum (exact under stable argsort
// tie-break) = sum(v > t) + (k - count(v > t)) * t.
// LDS usage: 67200 + 8192 + scalars ~= 75.5 KB (< 320 KB/WGP).
// ---------------------------------------------------------------------------
__global__ __launch_bounds__(256) void k_negmine(
    const float* __restrict__ lossc, Acc* __restrict__ acc)
{
    __shared__ float    srow[NP];
    __shared__ unsigned hist[2048];
    __shared__ unsigned sBin, sK, sCntGt;
    __shared__ float    sSum[8];

    const int b = blockIdx.x, tid = threadIdx.x;
    const int np = acc->npos[b];
    long long k0 = (long long)NEGPOS * np;
    if (k0 > NP - 1) k0 = NP - 1;
    if (k0 <= 0) return;               // uniform per block
    unsigned k = (unsigned)k0;

    // stage row into LDS (async DMA), overlap with pass-1 histogram zeroing
    {
        const char* rowg = (const char*)(lossc + (size_t)b * NP);
        unsigned base = lds_offset_of((const void*)srow);
        for (int i = tid; i < NP4; i += 256)
            async_load16(base + i * 16, rowg + (size_t)i * 16);
    }
    for (int i = tid; i < 2048; i += 256) hist[i] = 0;
    wait_async0();
    __syncthreads();

    const float4* row4 = (const float4*)srow;

    // pass 1: top 11 bits
    for (int i = tid; i < NP4; i += 256) {
        float4 v = row4[i];
        atomicAdd(&hist[__float_as_uint(v.x) >> 21], 1u);
        atomicAdd(&hist[__float_as_uint(v.y) >> 21], 1u);
        atomicAdd(&hist[__float_as_uint(v.z) >> 21], 1u);
        atomicAdd(&hist[__float_as_uint(v.w) >> 21], 1u);
    }
    __syncthreads();
    if (tid == 0) {
        unsigned cum = 0, bin = 0, kk = k;
        for (int i = 2047; i >= 0; --i) {
            unsigned c = hist[i];
            if (cum + c >= kk) { bin = (unsigned)i; kk -= cum; break; }
            cum += c;
        }
        sBin = bin; sK = kk;
    }
    __syncthreads();
    unsigned p1 = sBin; k = sK;

    // pass 2: middle 11 bits
    for (int i = tid; i < 2048; i += 256) hist[i] = 0;
    __syncthreads();
    for (int i = tid; i < NP4; i += 256) {
        float4 v = row4[i];
        unsigned u;
        u = __float_as_uint(v.x); if ((u >> 21) == p1) atomicAdd(&hist[(u >> 10) & 2047u], 1u);
        u = __float_as_uint(v.y); if ((u >> 21) == p1) atomicAdd(&hist[(u >> 10) & 2047u], 1u);
        u = __float_as_uint(v.z); if ((u >> 21) == p1) atomicAdd(&hist[(u >> 10) & 2047u], 1u);
        u = __float_as_uint(v.w); if ((u >> 21) == p1) atomicAdd(&hist[(u >> 10) & 2047u], 1u);
    }
    __syncthreads();
    if (tid == 0) {
        unsigned cum = 0, bin = 0, kk = k;
        for (int i = 2047; i >= 0; --i) {
            unsigned c = hist[i];
            if (cum + c >= kk) { bin = (unsigned)i; kk -= cum; break; }
            cum += c;
        }
        sBin = bin; sK = kk;
    }
    __syncthreads();
    unsigned p2 = sBin; k = sK;
    const unsigned hi21 = (p1 << 11) | p2;

    // pass 3: low 10 bits
    for (int i = tid; i < 1024; i += 256) hist[i] = 0;
    __syncthreads();
    for (int i = tid; i < NP4; i += 256) {
        float4 v = row4[i];
        unsigned u;
        u = __float_as_uint(v.x); if ((u >> 10) == hi21) atomicAdd(&hist[u & 1023u], 1u);
        u = __float_as_uint(v.y); if ((u >> 10) == hi21) atomicAdd(&hist[u & 1023u], 1u);
        u = __float_as_uint(v.z); if ((u >> 10) == hi21) atomicAdd(&hist[u & 1023u], 1u);
        u = __float_as_uint(v.w); if ((u >> 10) == hi21) atomicAdd(&hist[u & 1023u], 1u);
    }
    __syncthreads();
    if (tid == 0) {
        unsigned cum = 0, bin = 0;
        for (int i = 1023; i >= 0; --i) {
            unsigned c = hist[i];
            if (cum + c >= k) { bin = (unsigned)i; break; }
            cum += c;
        }
        sBin = bin;
        sCntGt = 0;
    }
    __syncthreads();
    const unsigned tbits = (p1 << 21) | (p2 << 10) | sBin;

    // pass 4: sum and count strictly above threshold
    float partial = 0.0f;
    unsigned cgt = 0;
    for (int i = tid; i < NP4; i += 256) {
        float4 v = row4[i];
        unsigned u;
        u = __float_as_uint(v.x); if (u > tbits) { partial += v.x; ++cgt; }
        u = __float_as_uint(v.y); if (u > tbits) { partial += v.y; ++cgt; }
        u = __float_as_uint(v.z); if (u > tbits) { partial += v.z; ++cgt; }
        u = __float_as_uint(v.w); if (u > tbits) { partial += v.w; ++cgt; }
    }
    float ws = wave_sum_wmma(partial);   // converged: EXEC all ones
    const int wave = tid >> 5, lane = tid & 31;
    if (lane == 0) sSum[wave] = ws;
    if (cgt) atomicAdd(&sCntGt, cgt);
    __syncthreads();

    if (tid == 0) {
        float sg = 0.f;
        #pragma unroll
        for (int w = 0; w < 8; ++w) sg += sSum[w];
        float t = __uint_as_float(tbits);
        float neg = sg + (float)(k0 - (long long)sCntGt) * t;
        atomicAdd(&acc->loss_c, neg);
    }
}

// ---------------------------------------------------------------------------
// Finalize: out = (loss_l/N, loss_c/N, loss_landm/N1)
// ---------------------------------------------------------------------------
__global__ void k_final(const Acc* __restrict__ acc, float* __restrict__ out) {
    if (blockIdx.x == 0 && threadIdx.x == 0) {
        float N  = fmaxf((float)acc->npos_total,   1.0f);
        float N1 = fmaxf((float)acc->nposlm_total, 1.0f);
        out[0] = acc->loss_l     / N;
        out[1] = acc->loss_c     / N;
        out[2] = acc->loss_landm / N1;
    }
}

extern "C" void kernel_launch(void* const* d_in, const int* in_sizes, int n_in,
                              void* d_out, int out_size, void* d_ws, size_t ws_size,
                              hipStream_t stream) {
    const float* loc_data   = (const float*)d_in[0];
    const float* conf_data  = (const float*)d_in[1];
    const float* landm_data = (const float*)d_in[2];
    const float* priors     = (const float*)d_in[3];
    const float* targets    = (const float*)d_in[4];

    char* ws = (char*)d_ws;
    size_t off = 0;
    auto carve = [&](size_t bytes) -> void* {
        void* ptr = (void*)(ws + off);
        off += (bytes + 255) & ~(size_t)255;
        return ptr;
    };
    float*              bto      = (float*)carve(sizeof(float) * NB * NP);
    int*                bti      = (int*)  carve(sizeof(int)   * NB * NP);
    int*                assigned = (int*)  carve(sizeof(int)   * NB * NP);
    float*              lossc    = (float*)carve(sizeof(float) * NB * NP);
    unsigned long long* bp       = (unsigned long long*)carve(sizeof(unsigned long long) * NB * NG);
    int*                anyvalid = (int*)  carve(sizeof(int) * NB);
    Acc*                acc      = (Acc*)  carve(sizeof(Acc));
    (void)ws_size; (void)in_sizes; (void)n_in; (void)out_size;

    const int ptiles = (NP + 255) / 256;

    k_init<<<dim3((NB * NG + 255) / 256), 256, 0, stream>>>(bp, acc);
    k_match<<<dim3(NB, ptiles), 256, 0, stream>>>(priors, targets, bto, bti, assigned, bp);
    k_bestprior<<<dim3(NB), 32, 0, stream>>>(bp, bto, assigned, anyvalid);
    k_encode_loss<<<dim3(NB, ptiles), 256, 0, stream>>>(
        loc_data, conf_data, landm_data, priors, targets,
        bto, bti, assigned, anyvalid, lossc, acc);
    k_negmine<<<dim3(NB), 256, 0, stream>>>(lossc, acc);
    k_final<<<1, 1, 0, stream>>>(acc, (float*)d_out);
}